// NormalConsistencyLoss_46420006536055
// MI455X (gfx1250) — compile-verified
//
#include <hip/hip_runtime.h>

// ---------------------------------------------------------------------------
// Normal-consistency loss over a triangulated 1024x1024 grid mesh, B=4.
// Memory-bound gather + reduction. CDNA5 (gfx1250, wave32) specifics used:
//   * V_WMMA_F32_16X16X4_F32 as an exact fp32 wave-wide sum (A = ones)
//   * one float64 global atomic per wave32
//   * global_prefetch for next grid-stride index loads
// ---------------------------------------------------------------------------

typedef __attribute__((ext_vector_type(2))) float v2f;
typedef __attribute__((ext_vector_type(8))) float v8f;

#define GRID_G 1024
#define BATCH  4

__global__ void nc_init_kernel(double* acc) {
    if (blockIdx.x == 0 && threadIdx.x == 0) acc[0] = 0.0;
}

__global__ __launch_bounds__(256) void nc_main_kernel(
    const float* __restrict__ x,      // (B, G*G, 3)
    const int*   __restrict__ ev0,    // (E,)
    const int*   __restrict__ ev1,
    const int*   __restrict__ oa,
    const int*   __restrict__ ob,
    int E,
    double* __restrict__ acc)
{
    const int stride = gridDim.x * blockDim.x;
    const int tid    = blockIdx.x * blockDim.x + threadIdx.x;
    const int N3     = GRID_G * GRID_G * 3;

    // Two fp32 partial accumulators per lane -> 64 values per wave,
    // exactly one 16x4 WMMA B operand (2 VGPRs x 32 lanes).
    float acc0 = 0.0f, acc1 = 0.0f;

    for (int e = tid; e < E; e += stride) {
        // Prefetch next iteration's index data (speculative, safe OOB).
        __builtin_prefetch(&ev0[e + stride], 0, 0);
        __builtin_prefetch(&ev1[e + stride], 0, 0);
        __builtin_prefetch(&oa [e + stride], 0, 0);
        __builtin_prefetch(&ob [e + stride], 0, 0);

        const int i0 = ev0[e] * 3;
        const int i1 = ev1[e] * 3;
        const int ia = oa[e]  * 3;
        const int ib = ob[e]  * 3;

#pragma unroll
        for (int b = 0; b < BATCH; ++b) {
            const float* __restrict__ xb = x + b * N3;

            const float p0x = xb[i0 + 0], p0y = xb[i0 + 1], p0z = xb[i0 + 2];
            const float p1x = xb[i1 + 0], p1y = xb[i1 + 1], p1z = xb[i1 + 2];
            const float pax = xb[ia + 0], pay = xb[ia + 1], paz = xb[ia + 2];
            const float pbx = xb[ib + 0], pby = xb[ib + 1], pbz = xb[ib + 2];

            // edge and opposite-vertex vectors
            const float ex = p1x - p0x, ey = p1y - p0y, ez = p1z - p0z;
            const float ax = pax - p0x, ay = pay - p0y, az = paz - p0z;
            const float bx = pbx - p0x, by = pby - p0y, bz = pbz - p0z;

            // n0 = cross(e, a)
            const float n0x = ey * az - ez * ay;
            const float n0y = ez * ax - ex * az;
            const float n0z = ex * ay - ey * ax;
            // n1 = -cross(e, b) = cross(b, e)
            const float n1x = by * ez - bz * ey;
            const float n1y = bz * ex - bx * ez;
            const float n1z = bx * ey - by * ex;

            const float dot = n0x * n1x + n0y * n1y + n0z * n1z;
            const float l0  = sqrtf(n0x * n0x + n0y * n0y + n0z * n0z);
            const float l1  = sqrtf(n1x * n1x + n1y * n1y + n1z * n1z);
            const float den = fmaxf(l0, 1e-8f) * fmaxf(l1, 1e-8f);
            const float loss = 1.0f - dot / den;

            if (b < 2) acc0 += loss; else acc1 += loss;
        }
    }

    // ---- Wave-wide exact fp32 reduction via WMMA ---------------------------
    // D = A(16x4, all ones) * B(4x16) + 0  =>  D[m][n] = sum_k B[k][n].
    // The 64 B elements are exactly {acc0,acc1} across the 32 lanes, each
    // mapped to a unique (k,n); an all-ones A makes every element contribute
    // once per D row, so sum over D VGPR0 (rows 0 and 8, both = column sums)
    // across all 32 lanes equals 2 * wave_total, independent of the layout.
    v2f Aones; Aones.x = 1.0f; Aones.y = 1.0f;
    v2f Bvals; Bvals.x = acc0; Bvals.y = acc1;
    v8f Czero = {};
    v8f D = __builtin_amdgcn_wmma_f32_16x16x4_f32(
        /*neg_a=*/false, Aones, /*neg_b=*/false, Bvals,
        /*c_mod=*/(short)0, Czero, /*reuse_a=*/false, /*reuse_b=*/false);

    float wsum = D[0];
#pragma unroll
    for (int off = 16; off >= 1; off >>= 1)
        wsum += __shfl_xor(wsum, off, 32);
    wsum *= 0.5f;  // rows 0 and 8 both present -> counted twice

    if ((threadIdx.x & 31) == 0)
        atomicAdd(acc, (double)wsum);  // global_atomic_add_f64, 1 per wave32
}

__global__ void nc_final_kernel(const double* __restrict__ acc,
                                float* __restrict__ out,
                                long long count) {
    if (blockIdx.x == 0 && threadIdx.x == 0)
        out[0] = (float)(acc[0] / (double)count);
}

extern "C" void kernel_launch(void* const* d_in, const int* in_sizes, int n_in,
                              void* d_out, int out_size, void* d_ws, size_t ws_size,
                              hipStream_t stream) {
    const float* x   = (const float*)d_in[0];
    const int*   ev0 = (const int*)d_in[1];
    const int*   ev1 = (const int*)d_in[2];
    const int*   oa  = (const int*)d_in[3];
    const int*   ob  = (const int*)d_in[4];
    const int    E   = in_sizes[1];

    double* acc = (double*)d_ws;
    float*  out = (float*)d_out;

    nc_init_kernel<<<1, 32, 0, stream>>>(acc);

    const int threads = 256;
    const int blocks  = 2048;  // ~6 edges/thread; fills the device many times over
    nc_main_kernel<<<blocks, threads, 0, stream>>>(x, ev0, ev1, oa, ob, E, acc);

    const long long count = (long long)BATCH * (long long)E;
    nc_final_kernel<<<1, 32, 0, stream>>>(acc, out, count);
}